// MentalHealthGNN_9345848836371
// MI455X (gfx1250) — compile-verified
//
#include <hip/hip_runtime.h>
#include <math.h>

namespace {

constexpr int NMOD   = 4;
constexpr int NNODE  = 10000;
constexpr int NEDGE  = 160000;
constexpr int E2     = NEDGE + NNODE;   // edges + self loops
constexpr int NGR    = 32;
constexpr int FIN    = 128;
constexpr int KIN    = 144;             // FIN + TDIM
constexpr int TD     = 16;
constexpr int H1C    = 256;             // HEADS*HID
constexpr int HIDC   = 64;
constexpr int NH1    = 4;               // heads layer 1
constexpr int EMB    = 32;
constexpr int MT     = 625;             // 10000/16 row tiles

typedef __attribute__((ext_vector_type(2))) float v2f;
typedef __attribute__((ext_vector_type(8))) float v8f;

struct HeadP { const float* w1; const float* b1; const float* w2; const float* b2; int odim; int ooff; };

__device__ __forceinline__ float warp_sum(float v) {
#pragma unroll
  for (int o = 16; o > 0; o >>= 1) v += __shfl_xor(v, o, 32);
  return v;
}

__device__ __forceinline__ void atomicMaxF(float* addr, float val) {
  int* ai = (int*)addr;
  int old = *ai;
  while (__int_as_float(old) < val) {
    int assumed = old;
    old = atomicCAS(ai, assumed, __float_as_int(val));
    if (old == assumed) break;
  }
}

__global__ void k_fill(float* p, float v, int n) {
  int i = blockIdx.x * blockDim.x + threadIdx.x;
  if (i < n) p[i] = v;
}

// ---- time encoding: scatter sin/cos features by src node --------------------
__global__ void k_time_encode(const float* __restrict__ etime, const int* __restrict__ ei,
                              const float* __restrict__ te_w, const float* __restrict__ te_b,
                              float* __restrict__ tsum, float* __restrict__ cnt) {
  long idx = (long)blockIdx.x * blockDim.x + threadIdx.x;
  if (idx >= (long)NMOD * NEDGE) return;
  int m = (int)(idx / NEDGE), e = (int)(idx % NEDGE);
  int src = ei[(m * 2 + 0) * NEDGE + e];
  float t = etime[(long)m * NEDGE + e];
  atomicAdd(&cnt[m * NNODE + src], 1.0f);
  float* dstp = &tsum[((long)(m * NNODE + src)) * TD];
#pragma unroll
  for (int j = 0; j < TD; ++j) {
    float tp = t * te_w[m * TD + j] + te_b[m * TD + j];
    float v = (j & 1) ? cosf(tp) : sinf(tp);
    atomicAdd(&dstp[j], v);
  }
}

__global__ void k_time_fin(float* __restrict__ tsum, const float* __restrict__ cnt) {
  long idx = (long)blockIdx.x * blockDim.x + threadIdx.x;
  if (idx >= (long)NMOD * NNODE * TD) return;
  tsum[idx] = tsum[idx] / (cnt[idx / TD] + 1.0f);
}

// ---- input projection GEMM with concat [x | node_time], fp32 WMMA ----------
__global__ void k_gemm_input(const float* __restrict__ x, const float* __restrict__ ntm,
                             const float* __restrict__ W, const float* __restrict__ bias,
                             float* __restrict__ C) {
  int gid = blockIdx.x;                       // NMOD * 625 * 8
  int mod = gid / (MT * 8);
  int rem = gid % (MT * 8);
  int mt = rem / 8, nt = rem % 8;
  const float* xm = x + (long)mod * NNODE * FIN;
  const float* nm = ntm + (long)mod * NNODE * TD;
  const float* Wm = W + (long)mod * FIN * KIN;
  const float* bm = bias + mod * FIN;
  float* Cm = C + (long)mod * NNODE * FIN;
  int lane = threadIdx.x, half = lane >> 4, l16 = lane & 15;
  int arow = mt * 16 + l16;
  int bcol = nt * 16 + l16;
  v8f c = {};
  for (int k0 = 0; k0 < KIN; k0 += 4) {
    int kk = k0 + 2 * half;
    v2f a, b;
    if (kk < FIN) { a.x = xm[(long)arow * FIN + kk]; a.y = xm[(long)arow * FIN + kk + 1]; }
    else          { a.x = nm[arow * TD + (kk - FIN)]; a.y = nm[arow * TD + (kk - FIN) + 1]; }
    b.x = Wm[(long)bcol * KIN + kk];
    b.y = Wm[(long)bcol * KIN + kk + 1];
    c = __builtin_amdgcn_wmma_f32_16x16x4_f32(false, a, false, b, (short)0, c, false, false);
  }
#pragma unroll
  for (int v = 0; v < 8; ++v) {
    int crow = mt * 16 + half * 8 + v;
    Cm[(long)crow * FIN + bcol] = c[v] + bm[bcol];
  }
}

// ---- generic C = A(N x K) @ W(Nout x K)^T, per-modality, fp32 WMMA ---------
__global__ void k_gemm(const float* __restrict__ A, const float* __restrict__ W,
                       float* __restrict__ C, int K, int Nout, int ntiles,
                       long aStride, long wStride, long cStride) {
  int gid = blockIdx.x;
  int tilesPer = MT * ntiles;
  int mod = gid / tilesPer;
  int rem = gid % tilesPer;
  int mt = rem / ntiles, nt = rem % ntiles;
  const float* Am = A + (long)mod * aStride;
  const float* Wm = W + (long)mod * wStride;
  float* Cm = C + (long)mod * cStride;
  int lane = threadIdx.x, half = lane >> 4, l16 = lane & 15;
  int arow = mt * 16 + l16;
  int bcol = nt * 16 + l16;
  v8f c = {};
  for (int k0 = 0; k0 < K; k0 += 4) {
    int kk = k0 + 2 * half;
    v2f a, b;
    a.x = Am[(long)arow * K + kk];
    a.y = Am[(long)arow * K + kk + 1];
    b.x = Wm[(long)bcol * K + kk];
    b.y = Wm[(long)bcol * K + kk + 1];
    c = __builtin_amdgcn_wmma_f32_16x16x4_f32(false, a, false, b, (short)0, c, false, false);
  }
#pragma unroll
  for (int v = 0; v < 8; ++v) {
    int crow = mt * 16 + half * 8 + v;
    Cm[(long)crow * Nout + bcol] = c[v];
  }
}

// ---- per-node attention scores s = <h, a> per head --------------------------
__global__ void k_scores(const float* __restrict__ h, const float* __restrict__ a_src,
                         const float* __restrict__ a_dst, float* __restrict__ ss,
                         float* __restrict__ sd, int H, int HC) {
  long idx = (long)blockIdx.x * blockDim.x + threadIdx.x;
  if (idx >= (long)NMOD * NNODE * H) return;
  int m = (int)(idx / ((long)NNODE * H));
  int r = (int)(idx % ((long)NNODE * H));
  int n = r / H, hh = r % H;
  const float* hp = h + ((long)(m * NNODE + n)) * (H * HC) + hh * HC;
  const float* as = a_src + (m * H + hh) * HC;
  const float* ad = a_dst + (m * H + hh) * HC;
  float vs = 0.f, vd = 0.f;
  for (int c = 0; c < HC; ++c) { float t = hp[c]; vs += t * as[c]; vd += t * ad[c]; }
  ss[idx] = vs;
  sd[idx] = vd;
}

// ---- edge softmax pass 1: leaky_relu + running max over dst -----------------
__global__ void k_edge_max(const float* __restrict__ ss, const float* __restrict__ sd,
                           const int* __restrict__ ei, float* __restrict__ dmax,
                           float* __restrict__ eout, int H) {
  long idx = (long)blockIdx.x * blockDim.x + threadIdx.x;
  if (idx >= (long)NMOD * E2 * H) return;
  int m = (int)(idx / ((long)E2 * H));
  int r = (int)(idx % ((long)E2 * H));
  int e = r / H, hh = r % H;
  int src, dst;
  if (e < NEDGE) { src = ei[(m * 2 + 0) * NEDGE + e]; dst = ei[(m * 2 + 1) * NEDGE + e]; }
  else           { src = dst = e - NEDGE; }
  float ev = ss[(long)(m * NNODE + src) * H + hh] + sd[(long)(m * NNODE + dst) * H + hh];
  ev = ev > 0.f ? ev : 0.2f * ev;
  eout[idx] = ev;
  atomicMaxF(&dmax[(long)(m * NNODE + dst) * H + hh], ev);
}

// ---- edge softmax pass 2: exp(e - max), accumulate denominator --------------
__global__ void k_edge_exp(const int* __restrict__ ei, const float* __restrict__ dmax,
                           float* __restrict__ eout, float* __restrict__ den, int H) {
  long idx = (long)blockIdx.x * blockDim.x + threadIdx.x;
  if (idx >= (long)NMOD * E2 * H) return;
  int m = (int)(idx / ((long)E2 * H));
  int r = (int)(idx % ((long)E2 * H));
  int e = r / H, hh = r % H;
  int dst = (e < NEDGE) ? ei[(m * 2 + 1) * NEDGE + e] : (e - NEDGE);
  float ex = expf(eout[idx] - dmax[(long)(m * NNODE + dst) * H + hh]);
  eout[idx] = ex;
  atomicAdd(&den[(long)(m * NNODE + dst) * H + hh], ex);
}

// ---- edge aggregation: agg[dst] += alpha * h[src] ---------------------------
__global__ void k_edge_agg(const float* __restrict__ hsrc, const float* __restrict__ ex,
                           const float* __restrict__ den, const int* __restrict__ ei,
                           float* __restrict__ agg, int C, int H, int HC) {
  long idx = (long)blockIdx.x * blockDim.x + threadIdx.x;
  if (idx >= (long)NMOD * E2 * C) return;
  int m = (int)(idx / ((long)E2 * C));
  long r = idx % ((long)E2 * C);
  int e = (int)(r / C), c = (int)(r % C);
  int src, dst;
  if (e < NEDGE) { src = ei[(m * 2 + 0) * NEDGE + e]; dst = ei[(m * 2 + 1) * NEDGE + e]; }
  else           { src = dst = e - NEDGE; }
  int hh = c / HC;
  float alpha = ex[((long)m * E2 + e) * H + hh] /
                (den[(long)(m * NNODE + dst) * H + hh] + 1e-16f);
  atomicAdd(&agg[((long)(m * NNODE + dst)) * C + c],
            hsrc[((long)(m * NNODE + src)) * C + c] * alpha);
}

// ---- bias (+optional ELU) + LayerNorm, one wave per node --------------------
__global__ void k_post_ln(float* __restrict__ agg, const float* __restrict__ bias,
                          const float* __restrict__ g, const float* __restrict__ bln,
                          int C, int cpl, int doElu) {
  int mn = blockIdx.x;                 // NMOD*NNODE
  int m = mn / NNODE;
  int lane = threadIdx.x;
  float* row = agg + (long)mn * C;
  const float* bb = bias + m * C;
  const float* gg = g + m * C;
  const float* nb = bln + m * C;
  float v[8];
  float s = 0.f;
  for (int i = 0; i < cpl; ++i) {
    int c = i * 32 + lane;
    float t = row[c] + bb[c];
    if (doElu) t = t > 0.f ? t : (expf(t) - 1.f);
    v[i] = t; s += t;
  }
  float mean = warp_sum(s) / (float)C;
  float s2 = 0.f;
  for (int i = 0; i < cpl; ++i) { float d = v[i] - mean; s2 += d * d; }
  float var = warp_sum(s2) / (float)C;
  float inv = rsqrtf(var + 1e-5f);
  for (int i = 0; i < cpl; ++i) {
    int c = i * 32 + lane;
    row[c] = (v[i] - mean) * inv * gg[c] + nb[c];
  }
}

// ---- graph pooling ----------------------------------------------------------
__global__ void k_gcnt(const int* __restrict__ batch, float* __restrict__ gcnt) {
  int i = blockIdx.x * blockDim.x + threadIdx.x;
  if (i < NNODE) atomicAdd(&gcnt[batch[i]], 1.0f);
}

__global__ void k_pool(const float* __restrict__ h, const int* __restrict__ batch,
                       float* __restrict__ pooled) {
  long idx = (long)blockIdx.x * blockDim.x + threadIdx.x;
  if (idx >= (long)NMOD * NNODE * EMB) return;
  int m = (int)(idx / (NNODE * EMB));
  int r = (int)(idx % (NNODE * EMB));
  int n = r / EMB, c = r % EMB;
  int g = batch[n];
  atomicAdd(&pooled[(m * NGR + g) * EMB + c], h[((long)(m * NNODE + n)) * EMB + c]);
}

// ---- fusion attention + MLP heads: 32 blocks (graphs) x 32 threads ----------
__device__ void run_head(const float* fused_s, const HeadP hp, float* s_h1,
                         float* out, int b, int t) {
  if (t < 16) {
    float a = hp.b1[t];
    for (int j = 0; j < 32; ++j) a += hp.w1[t * 32 + j] * fused_s[j];
    s_h1[t] = fmaxf(a, 0.f);
  }
  __syncthreads();
  if (t < hp.odim) {
    float a = hp.b2[t];
    for (int k = 0; k < 16; ++k) a += hp.w2[t * 16 + k] * s_h1[k];
    out[b * 12 + hp.ooff + t] = a;
  }
  __syncthreads();
}

__global__ void k_fusion(const float* __restrict__ pooled, const float* __restrict__ gcnt,
                         const float* __restrict__ mod_embed,
                         const float* __restrict__ wq, const float* __restrict__ bq,
                         const float* __restrict__ wk, const float* __restrict__ bk,
                         const float* __restrict__ wv, const float* __restrict__ bv,
                         const float* __restrict__ wo, const float* __restrict__ bo,
                         const float* __restrict__ ow, const float* __restrict__ ob,
                         const float* __restrict__ lg, const float* __restrict__ lb,
                         HeadP hA, HeadP hB, HeadP hC, HeadP hD,
                         float* __restrict__ out) {
  __shared__ float s_q[32], s_xe[4][32], s_Q[32], s_K[4][32], s_V[4][32];
  __shared__ float s_sc[4][4], s_a[4][4], s_o[32], s_attn[32], s_fused[32], s_h1[16];
  int b = blockIdx.x, t = threadIdx.x;
  float cb = fmaxf(gcnt[b], 1.0f);
  float xe[4], q = 0.f;
#pragma unroll
  for (int m = 0; m < 4; ++m) {
    xe[m] = pooled[(m * NGR + b) * EMB + t] / cb + mod_embed[m * EMB + t];
    q += xe[m];
    s_xe[m][t] = xe[m];
  }
  s_q[t] = q * 0.25f;
  __syncthreads();
  float Qv = bq[t];
  for (int j = 0; j < 32; ++j) Qv += wq[t * 32 + j] * s_q[j];
  s_Q[t] = Qv;
#pragma unroll
  for (int m = 0; m < 4; ++m) {
    float Kv = bk[t], Vv = bv[t];
    for (int j = 0; j < 32; ++j) {
      Kv += wk[t * 32 + j] * s_xe[m][j];
      Vv += wv[t * 32 + j] * s_xe[m][j];
    }
    s_K[m][t] = Kv; s_V[m][t] = Vv;
  }
  __syncthreads();
  if (t < 16) {
    int h = t >> 2, m = t & 3;
    float s = 0.f;
    for (int d = 0; d < 8; ++d) s += s_Q[h * 8 + d] * s_K[m][h * 8 + d];
    s_sc[h][m] = s * 0.35355339059327373f;   // 1/sqrt(8)
  }
  __syncthreads();
  if (t < 4) {
    float mx = -1e30f;
    for (int m = 0; m < 4; ++m) mx = fmaxf(mx, s_sc[t][m]);
    float e[4], su = 0.f;
    for (int m = 0; m < 4; ++m) { e[m] = expf(s_sc[t][m] - mx); su += e[m]; }
    for (int m = 0; m < 4; ++m) s_a[t][m] = e[m] / su;
  }
  __syncthreads();
  int hh = t >> 3;
  float o = 0.f;
  for (int m = 0; m < 4; ++m) o += s_a[hh][m] * s_V[m][t];
  s_o[t] = o;
  __syncthreads();
  float at = bo[t];
  for (int j = 0; j < 32; ++j) at += wo[t * 32 + j] * s_o[j];
  s_attn[t] = at;
  __syncthreads();
  float fp = ob[t];
  for (int j = 0; j < 32; ++j) fp += ow[t * 32 + j] * s_attn[j];
  float mean = warp_sum(fp) * (1.f / 32.f);
  float d = fp - mean;
  float var = warp_sum(d * d) * (1.f / 32.f);
  s_fused[t] = d * rsqrtf(var + 1e-5f) * lg[t] + lb[t];
  __syncthreads();
  run_head(s_fused, hA, s_h1, out, b, t);
  run_head(s_fused, hB, s_h1, out, b, t);
  run_head(s_fused, hC, s_h1, out, b, t);
  run_head(s_fused, hD, s_h1, out, b, t);
}

} // namespace

extern "C" void kernel_launch(void* const* d_in, const int* in_sizes, int n_in,
                              void* d_out, int out_size, void* d_ws, size_t ws_size,
                              hipStream_t stream) {
  (void)in_sizes; (void)n_in; (void)out_size; (void)ws_size;
  const float* x     = (const float*)d_in[0];
  const float* etime = (const float*)d_in[1];
  const int*   ei    = (const int*)d_in[2];
  const int*   batch = (const int*)d_in[3];
  const float* te_w  = (const float*)d_in[4];
  const float* te_b  = (const float*)d_in[5];
  const float* inp_w = (const float*)d_in[6];
  const float* inp_b = (const float*)d_in[7];
  const float* g1_w  = (const float*)d_in[8];
  const float* g1_as = (const float*)d_in[9];
  const float* g1_ad = (const float*)d_in[10];
  const float* g1_b  = (const float*)d_in[11];
  const float* g2_w  = (const float*)d_in[12];
  const float* g2_as = (const float*)d_in[13];
  const float* g2_ad = (const float*)d_in[14];
  const float* g2_b  = (const float*)d_in[15];
  const float* n1_g  = (const float*)d_in[16];
  const float* n1_b  = (const float*)d_in[17];
  const float* n2_g  = (const float*)d_in[18];
  const float* n2_b  = (const float*)d_in[19];
  const float* mod_e = (const float*)d_in[20];
  const float* wq = (const float*)d_in[21]; const float* bq = (const float*)d_in[22];
  const float* wk = (const float*)d_in[23]; const float* bk = (const float*)d_in[24];
  const float* wv = (const float*)d_in[25]; const float* bv = (const float*)d_in[26];
  const float* wo = (const float*)d_in[27]; const float* bo = (const float*)d_in[28];
  const float* ow = (const float*)d_in[29]; const float* ob = (const float*)d_in[30];
  const float* lg = (const float*)d_in[31]; const float* lb = (const float*)d_in[32];
  HeadP hA = {(const float*)d_in[33], (const float*)d_in[34], (const float*)d_in[35], (const float*)d_in[36], 1, 0};
  HeadP hB = {(const float*)d_in[37], (const float*)d_in[38], (const float*)d_in[39], (const float*)d_in[40], 5, 1};
  HeadP hC = {(const float*)d_in[41], (const float*)d_in[42], (const float*)d_in[43], (const float*)d_in[44], 5, 6};
  HeadP hD = {(const float*)d_in[45], (const float*)d_in[46], (const float*)d_in[47], (const float*)d_in[48], 1, 11};
  float* out = (float*)d_out;

  float* ws = (float*)d_ws;
  size_t o_tsum = 0;
  size_t o_cnt  = o_tsum + (size_t)NMOD * NNODE * TD;
  size_t o_xm   = o_cnt  + (size_t)NMOD * NNODE;
  size_t o_h1   = o_xm   + (size_t)NMOD * NNODE * FIN;
  size_t o_s1s  = o_h1   + (size_t)NMOD * NNODE * H1C;
  size_t o_s1d  = o_s1s  + (size_t)NMOD * NNODE * NH1;
  size_t o_dm1  = o_s1d  + (size_t)NMOD * NNODE * NH1;
  size_t o_dn1  = o_dm1  + (size_t)NMOD * NNODE * NH1;
  size_t o_ex1  = o_dn1  + (size_t)NMOD * NNODE * NH1;
  size_t o_ag1  = o_ex1  + (size_t)NMOD * E2 * NH1;
  size_t o_h2   = o_ag1  + (size_t)NMOD * NNODE * H1C;
  size_t o_s2s  = o_h2   + (size_t)NMOD * NNODE * EMB;
  size_t o_s2d  = o_s2s  + (size_t)NMOD * NNODE;
  size_t o_dm2  = o_s2d  + (size_t)NMOD * NNODE;
  size_t o_dn2  = o_dm2  + (size_t)NMOD * NNODE;
  size_t o_ex2  = o_dn2  + (size_t)NMOD * NNODE;
  size_t o_ag2  = o_ex2  + (size_t)NMOD * E2;
  size_t o_pool = o_ag2  + (size_t)NMOD * NNODE * EMB;
  size_t o_gcnt = o_pool + (size_t)NMOD * NGR * EMB;
  size_t total  = o_gcnt + NGR;

  hipMemsetAsync(d_ws, 0, total * sizeof(float), stream);
  {
    int n1 = NMOD * NNODE * NH1;
    k_fill<<<(n1 + 255) / 256, 256, 0, stream>>>(ws + o_dm1, -1e30f, n1);
    int n2 = NMOD * NNODE;
    k_fill<<<(n2 + 255) / 256, 256, 0, stream>>>(ws + o_dm2, -1e30f, n2);
  }

  // 1) time encoding scatter + finalize mean
  {
    long n = (long)NMOD * NEDGE;
    k_time_encode<<<(unsigned)((n + 255) / 256), 256, 0, stream>>>(etime, ei, te_w, te_b, ws + o_tsum, ws + o_cnt);
    long nf = (long)NMOD * NNODE * TD;
    k_time_fin<<<(unsigned)((nf + 255) / 256), 256, 0, stream>>>(ws + o_tsum, ws + o_cnt);
  }
  // 2) input GEMM (N x 144) @ (144 x 128) + bias, fp32 WMMA
  k_gemm_input<<<NMOD * MT * 8, 32, 0, stream>>>(x, ws + o_tsum, inp_w, inp_b, ws + o_xm);
  // 3) GAT1 projection GEMM (N x 128) @ (128 x 256)
  k_gemm<<<NMOD * MT * 16, 32, 0, stream>>>(ws + o_xm, g1_w, ws + o_h1, FIN, H1C, 16,
                                            (long)NNODE * FIN, (long)H1C * FIN, (long)NNODE * H1C);
  // 4) GAT1 edge softmax + aggregation
  {
    long ns = (long)NMOD * NNODE * NH1;
    k_scores<<<(unsigned)((ns + 255) / 256), 256, 0, stream>>>(ws + o_h1, g1_as, g1_ad,
                                                               ws + o_s1s, ws + o_s1d, NH1, HIDC);
    long nE = (long)NMOD * E2 * NH1;
    k_edge_max<<<(unsigned)((nE + 255) / 256), 256, 0, stream>>>(ws + o_s1s, ws + o_s1d, ei,
                                                                 ws + o_dm1, ws + o_ex1, NH1);
    k_edge_exp<<<(unsigned)((nE + 255) / 256), 256, 0, stream>>>(ei, ws + o_dm1, ws + o_ex1,
                                                                 ws + o_dn1, NH1);
    long nA = (long)NMOD * E2 * H1C;
    k_edge_agg<<<(unsigned)((nA + 255) / 256), 256, 0, stream>>>(ws + o_h1, ws + o_ex1, ws + o_dn1,
                                                                 ei, ws + o_ag1, H1C, NH1, HIDC);
  }
  // 5) bias + ELU + LayerNorm (in place on agg1)
  k_post_ln<<<NMOD * NNODE, 32, 0, stream>>>(ws + o_ag1, g1_b, n1_g, n1_b, H1C, H1C / 32, 1);
  // 6) GAT2 projection GEMM (N x 256) @ (256 x 32)
  k_gemm<<<NMOD * MT * 2, 32, 0, stream>>>(ws + o_ag1, g2_w, ws + o_h2, H1C, EMB, 2,
                                           (long)NNODE * H1C, (long)EMB * H1C, (long)NNODE * EMB);
  // 7) GAT2 edge softmax + aggregation (1 head)
  {
    long ns = (long)NMOD * NNODE;
    k_scores<<<(unsigned)((ns + 255) / 256), 256, 0, stream>>>(ws + o_h2, g2_as, g2_ad,
                                                               ws + o_s2s, ws + o_s2d, 1, EMB);
    long nE = (long)NMOD * E2;
    k_edge_max<<<(unsigned)((nE + 255) / 256), 256, 0, stream>>>(ws + o_s2s, ws + o_s2d, ei,
                                                                 ws + o_dm2, ws + o_ex2, 1);
    k_edge_exp<<<(unsigned)((nE + 255) / 256), 256, 0, stream>>>(ei, ws + o_dm2, ws + o_ex2,
                                                                 ws + o_dn2, 1);
    long nA = (long)NMOD * E2 * EMB;
    k_edge_agg<<<(unsigned)((nA + 255) / 256), 256, 0, stream>>>(ws + o_h2, ws + o_ex2, ws + o_dn2,
                                                                 ei, ws + o_ag2, EMB, 1, EMB);
  }
  // 8) bias + LayerNorm (no ELU, in place on agg2)
  k_post_ln<<<NMOD * NNODE, 32, 0, stream>>>(ws + o_ag2, g2_b, n2_g, n2_b, EMB, 1, 0);
  // 9) graph pooling
  k_gcnt<<<(NNODE + 255) / 256, 256, 0, stream>>>(batch, ws + o_gcnt);
  {
    long n = (long)NMOD * NNODE * EMB;
    k_pool<<<(unsigned)((n + 255) / 256), 256, 0, stream>>>(ws + o_ag2, batch, ws + o_pool);
  }
  // 10) fusion attention + heads
  k_fusion<<<NGR, 32, 0, stream>>>(ws + o_pool, ws + o_gcnt, mod_e,
                                   wq, bq, wk, bk, wv, bv, wo, bo, ow, ob, lg, lb,
                                   hA, hB, hC, hD, out);
}